// SpatialEncoder_69097433858240
// MI455X (gfx1250) — compile-verified
//
#include <hip/hip_runtime.h>

#define C_IN 64
#define HC   128          // HEADS * C_OUT
#define NEG_SLOPE 0.2f

typedef float v2f __attribute__((ext_vector_type(2)));
typedef float v8f __attribute__((ext_vector_type(8)));

// ---- monotonic uint key for float atomic max (handles negatives) ----
__device__ __forceinline__ unsigned fkey(float f) {
    unsigned u = __float_as_uint(f);
    return (u & 0x80000000u) ? ~u : (u | 0x80000000u);
}
__device__ __forceinline__ float funkey(unsigned k) {
    unsigned u = (k & 0x80000000u) ? (k & 0x7FFFFFFFu) : ~k;
    return __uint_as_float(u);
}

// ---- init: out = bias (broadcast), denom = 0, max-keys = 0 (== -inf key) ----
__global__ void init_kernel(float* __restrict__ out, const float* __restrict__ bias,
                            float* __restrict__ denom, unsigned* __restrict__ mkey, int T) {
    int idx = blockIdx.x * blockDim.x + threadIdx.x;
    if (idx < T * HC) out[idx] = bias[idx & (HC - 1)];
    if (idx < T * 2) { denom[idx] = 0.0f; mkey[idx] = 0u; }
}

// ---- fp32 WMMA projection: wave 0 -> xl = x*Wl, wave 1 -> xr = x*Wr ----
// One wave computes a 16-row x 128-col slab using V_WMMA_F32_16X16X4_F32.
__global__ void proj_kernel(const float* __restrict__ x,   // T x 64
                            const float* __restrict__ Wl,  // 64 x 128
                            const float* __restrict__ Wr,  // 64 x 128
                            float* __restrict__ xl,        // T x 128
                            float* __restrict__ xr) {
    const int wave = threadIdx.x >> 5;          // 0 or 1
    const int lane = threadIdx.x & 31;
    const int halfSel = lane >> 4;              // 0: lanes 0-15, 1: lanes 16-31
    const int l15 = lane & 15;
    const int row0 = blockIdx.x * 16;

    const float* __restrict__ W   = (wave == 0) ? Wl : Wr;
    float*       __restrict__ dst = (wave == 0) ? xl : xr;

    v8f acc[8] = {};                            // 8 N-tiles of 16 cols each

    #pragma unroll
    for (int k0 = 0; k0 < C_IN; k0 += 4) {
        // A (16x4 f32): lanes 0-15 carry K=k0,k0+1; lanes 16-31 carry K=k0+2,k0+3
        const float* ap = x + (size_t)(row0 + l15) * C_IN + k0 + 2 * halfSel;
        v2f a; a.x = ap[0]; a.y = ap[1];
        #pragma unroll
        for (int nt = 0; nt < 8; ++nt) {
            // B (4x16 f32): lane = column; same K split as A
            const float* bp = W + (size_t)(k0 + 2 * halfSel) * HC + nt * 16 + l15;
            v2f b; b.x = bp[0]; b.y = bp[HC];
            acc[nt] = __builtin_amdgcn_wmma_f32_16x16x4_f32(
                false, a, false, b, (short)0, acc[nt], false, false);
        }
    }
    // C/D layout: VGPR r -> row = row0 + r + 8*halfSel, col = nt*16 + l15
    #pragma unroll
    for (int nt = 0; nt < 8; ++nt)
        #pragma unroll
        for (int r = 0; r < 8; ++r)
            dst[(size_t)(row0 + r + 8 * halfSel) * HC + nt * 16 + l15] = acc[nt][r];
}

// ---- per-edge GATv2 score + segment max (one wave32 per edge) ----
__global__ void edge_score_kernel(const float* __restrict__ xl, const float* __restrict__ xr,
                                  const int* __restrict__ src, const int* __restrict__ dst,
                                  const float* __restrict__ att,
                                  float* __restrict__ score, unsigned* __restrict__ mkey,
                                  int E, int EP) {
    int e = blockIdx.x * (blockDim.x >> 5) + (threadIdx.x >> 5);
    if (e >= EP) return;
    const int lane = threadIdx.x & 31;
    int s, d;
    if (e < E) { s = src[e]; d = dst[e]; } else { s = e - E; d = e - E; }

    float4 a = ((const float4*)(xl + (size_t)s * HC))[lane];
    float4 b = ((const float4*)(xr + (size_t)d * HC))[lane];
    float4 w = ((const float4*)att)[lane];      // att flat (H*C = 128)

    float v, sum = 0.0f;
    v = a.x + b.x; v = (v > 0.0f) ? v : NEG_SLOPE * v; sum += v * w.x;
    v = a.y + b.y; v = (v > 0.0f) ? v : NEG_SLOPE * v; sum += v * w.y;
    v = a.z + b.z; v = (v > 0.0f) ? v : NEG_SLOPE * v; sum += v * w.z;
    v = a.w + b.w; v = (v > 0.0f) ? v : NEG_SLOPE * v; sum += v * w.w;

    // reduce within each 16-lane half (lanes 0-15 = head 0, 16-31 = head 1)
    sum += __shfl_down(sum, 8, 16);
    sum += __shfl_down(sum, 4, 16);
    sum += __shfl_down(sum, 2, 16);
    sum += __shfl_down(sum, 1, 16);

    if ((lane & 15) == 0) {
        int h = lane >> 4;
        score[(size_t)e * 2 + h] = sum;
        atomicMax(&mkey[(size_t)d * 2 + h], fkey(sum));
    }
}

// ---- exp(score - max) and segment-sum denominators (thread per edge,head) ----
__global__ void softmax_kernel(const int* __restrict__ dst, const unsigned* __restrict__ mkey,
                               float* __restrict__ score, float* __restrict__ denom,
                               int E, int EP) {
    int idx = blockIdx.x * blockDim.x + threadIdx.x;
    if (idx >= EP * 2) return;
    int e = idx >> 1, h = idx & 1;
    int d = (e < E) ? dst[e] : e - E;
    float m = funkey(mkey[(size_t)d * 2 + h]);
    float ea = __expf(score[idx] - m);
    score[idx] = ea;
    atomicAdd(&denom[(size_t)d * 2 + h], ea);
}

// ---- scatter-aggregate: out[dst] += alpha * xl[src] (one wave32 per edge) ----
__global__ void aggregate_kernel(const float* __restrict__ xl,
                                 const int* __restrict__ src, const int* __restrict__ dst,
                                 const float* __restrict__ score, const float* __restrict__ denom,
                                 float* __restrict__ out, int E, int EP) {
    int e = blockIdx.x * (blockDim.x >> 5) + (threadIdx.x >> 5);
    if (e >= EP) return;
    const int lane = threadIdx.x & 31;
    int s, d;
    if (e < E) { s = src[e]; d = dst[e]; } else { s = e - E; d = e - E; }
    const int h = lane >> 4;
    float alpha = score[(size_t)e * 2 + h] / denom[(size_t)d * 2 + h];
    float4 v = ((const float4*)(xl + (size_t)s * HC))[lane];
    float* o = out + (size_t)d * HC + lane * 4;
    atomicAdd(o + 0, alpha * v.x);
    atomicAdd(o + 1, alpha * v.y);
    atomicAdd(o + 2, alpha * v.z);
    atomicAdd(o + 3, alpha * v.w);
}

extern "C" void kernel_launch(void* const* d_in, const int* in_sizes, int n_in,
                              void* d_out, int out_size, void* d_ws, size_t ws_size,
                              hipStream_t stream) {
    const float* x    = (const float*)d_in[0];
    const int*   ei   = (const int*)d_in[1];
    const float* Wl   = (const float*)d_in[2];
    const float* Wr   = (const float*)d_in[3];
    const float* att  = (const float*)d_in[4];
    const float* bias = (const float*)d_in[5];
    float* out = (float*)d_out;

    const int T  = in_sizes[0] / C_IN;   // 131072
    const int E  = in_sizes[1] / 2;      // 1,000,000
    const int EP = E + T;                // edges + self loops
    const int* src = ei;
    const int* dst = ei + E;

    // workspace layout (all 16B aligned)
    char* ws = (char*)d_ws;
    float*    xl    = (float*)ws;    ws += (size_t)T * HC * sizeof(float);
    float*    xr    = (float*)ws;    ws += (size_t)T * HC * sizeof(float);
    float*    score = (float*)ws;    ws += (size_t)EP * 2 * sizeof(float);
    float*    denom = (float*)ws;    ws += (size_t)T * 2 * sizeof(float);
    unsigned* mkey  = (unsigned*)ws; ws += (size_t)T * 2 * sizeof(unsigned);
    (void)ws_size; (void)n_in; (void)out_size;

    // 1) out = bias, denom = 0, max-keys = 0
    {
        int total = T * HC;
        init_kernel<<<(total + 255) / 256, 256, 0, stream>>>(out, bias, denom, mkey, T);
    }
    // 2) fp32 WMMA projections (2 waves/block: xl and xr for one 16-row tile)
    proj_kernel<<<T / 16, 64, 0, stream>>>(x, Wl, Wr, xl, xr);
    // 3) edge scores + segment max (wave per edge)
    edge_score_kernel<<<(EP + 7) / 8, 256, 0, stream>>>(xl, xr, src, dst, att, score, mkey, E, EP);
    // 4) exp + segment sum
    {
        int total = EP * 2;
        softmax_kernel<<<(total + 255) / 256, 256, 0, stream>>>(dst, mkey, score, denom, E, EP);
    }
    // 5) normalize + scatter aggregate (wave per edge)
    aggregate_kernel<<<(EP + 7) / 8, 256, 0, stream>>>(xl, src, dst, score, denom, out, E, EP);
}